// TropixLinear_4140348473994
// MI455X (gfx1250) — compile-verified
//
#include <hip/hip_runtime.h>
#include <cstdint>

// Tropical (max-times) linear: out[n,m] = max_k(xp*wp) + max_k(xn*wn)
//                                       - max_k(xp*wn) - max_k(xn*wp) + bias[m]
// N = K = M = 512 (x:[N,K], W:[M,K], bias:[M], out:[N,M], all fp32)
//
// WMMA hardware implements the (+,*) semiring only; a max-reduction GEMM
// cannot use it. We instead exercise the CDNA5 async-memory path:
// double-buffered global_load_async_to_lds_b128 staging with partial
// s_wait_asynccnt (in-order ASYNCcnt completion), LDS-tiled, register-blocked
// wave32 VALU compute with v_max3-friendly max chains.

#define N_DIM 512
#define K_DIM 512
#define M_DIM 512

#define TILE 32             // output tile per 256-thread block
#define KC   64             // K chunk staged in LDS per stage
#define XS_STRIDE (KC + 4)  // 68 floats = 272B row stride: 16B-aligned, bank-conflict-free stride

typedef __attribute__((address_space(3))) void lds_void_t;

// Raw LDS byte offset of a generic pointer into __shared__ memory.
__device__ __forceinline__ unsigned lds_off(const void* p) {
    return (unsigned)(unsigned long long)(lds_void_t*)(p);
}

// Async copy 16 bytes: from the global address saddr + voff to the LDS offset in ldsa   (ASYNCcnt-tracked)
__device__ __forceinline__ void async_b128_to_lds(unsigned ldsa, unsigned voff, uint64_t saddr) {
    asm volatile("global_load_async_to_lds_b128 %0, %1, %2"
                 :
                 : "v"(ldsa), "v"(voff), "s"(saddr)
                 : "memory");
}

__device__ __forceinline__ void wait_async_le(int n) {
    // ASYNCcnt ops complete in order: waiting "<= 4" after issuing 4 new copies
    // guarantees the previous 4 (the chunk we are about to consume) landed.
    if (n == 0) asm volatile("s_wait_asynccnt 0" ::: "memory");
    else        asm volatile("s_wait_asynccnt 4" ::: "memory");
}

__global__ __launch_bounds__(256)
void TropixLinear_4140348473994_kernel(const float* __restrict__ x,
                                       const float* __restrict__ w,
                                       const float* __restrict__ bias,
                                       float* __restrict__ out) {
    __shared__ __align__(16) float xs[2][TILE * XS_STRIDE];
    __shared__ __align__(16) float ws[2][TILE * XS_STRIDE];

    const int tid = threadIdx.x;
    const int rowBase = blockIdx.y * TILE;   // n
    const int colBase = blockIdx.x * TILE;   // m

    // 2x2 micro-tile per thread
    const int ty = tid >> 4;                 // 0..15
    const int tx = tid & 15;                 // 0..15
    const int r0 = ty * 2;
    const int c0 = tx * 2;

    // ---- async staging addressing: one float4 per thread per issue ----
    // tile = 32 rows x 64 floats = 512 float4; 256 threads -> 2 issues/tile
    const int f4row = tid >> 4;
    const int f4col = tid & 15;
    const unsigned gvoff0 = (unsigned)((f4row        * K_DIM + f4col * 4) * 4);
    const unsigned gvoff1 = (unsigned)(((f4row + 16) * K_DIM + f4col * 4) * 4);

    unsigned xlds0[2], xlds1[2], wlds0[2], wlds1[2];
    #pragma unroll
    for (int b = 0; b < 2; ++b) {
        xlds0[b] = lds_off(&xs[b][(f4row)      * XS_STRIDE + f4col * 4]);
        xlds1[b] = lds_off(&xs[b][(f4row + 16) * XS_STRIDE + f4col * 4]);
        wlds0[b] = lds_off(&ws[b][(f4row)      * XS_STRIDE + f4col * 4]);
        wlds1[b] = lds_off(&ws[b][(f4row + 16) * XS_STRIDE + f4col * 4]);
    }

    const float* xbase = x + (size_t)rowBase * K_DIM;
    const float* wbase = w + (size_t)colBase * K_DIM;

    // four max-accumulators per output (reduced values all >= 0 -> init 0 exact)
    float acc_pp[2][2] = {{0.f, 0.f}, {0.f, 0.f}};
    float acc_nn[2][2] = {{0.f, 0.f}, {0.f, 0.f}};
    float acc_pn[2][2] = {{0.f, 0.f}, {0.f, 0.f}};
    float acc_np[2][2] = {{0.f, 0.f}, {0.f, 0.f}};

    // prologue: stage chunk 0 into buffer 0
    {
        const uint64_t xsad = (uint64_t)(uintptr_t)(xbase);
        const uint64_t wsad = (uint64_t)(uintptr_t)(wbase);
        async_b128_to_lds(xlds0[0], gvoff0, xsad);
        async_b128_to_lds(xlds1[0], gvoff1, xsad);
        async_b128_to_lds(wlds0[0], gvoff0, wsad);
        async_b128_to_lds(wlds1[0], gvoff1, wsad);
    }

    const int NCHUNK = K_DIM / KC;           // 8
    for (int i = 0; i < NCHUNK; ++i) {
        const int cur = i & 1;
        const int nxt = cur ^ 1;

        // stage chunk i+1 into the other buffer (safe: that buffer was last
        // consumed in iteration i-1 and released by its trailing barrier)
        if (i + 1 < NCHUNK) {
            const int k0n = (i + 1) * KC;
            const uint64_t xsad = (uint64_t)(uintptr_t)(xbase + k0n);
            const uint64_t wsad = (uint64_t)(uintptr_t)(wbase + k0n);
            async_b128_to_lds(xlds0[nxt], gvoff0, xsad);
            async_b128_to_lds(xlds1[nxt], gvoff1, xsad);
            async_b128_to_lds(wlds0[nxt], gvoff0, wsad);
            async_b128_to_lds(wlds1[nxt], gvoff1, wsad);
            wait_async_le(4);                // chunk i landed; i+1 still in flight
        } else {
            wait_async_le(0);
        }
        __syncthreads();                     // all waves' copies of chunk i visible

        const float* xr0 = &xs[cur][(r0)     * XS_STRIDE];
        const float* xr1 = &xs[cur][(r0 + 1) * XS_STRIDE];
        const float* wr0 = &ws[cur][(c0)     * XS_STRIDE];
        const float* wr1 = &ws[cur][(c0 + 1) * XS_STRIDE];

        #pragma unroll 2
        for (int kk = 0; kk < KC; kk += 4) {
            const float4 xv0 = *(const float4*)(xr0 + kk);
            const float4 xv1 = *(const float4*)(xr1 + kk);
            const float4 wv0 = *(const float4*)(wr0 + kk);
            const float4 wv1 = *(const float4*)(wr1 + kk);

            float xp[2][4], xn[2][4], wp[2][4], wn[2][4];
            {
                const float xe0[4] = {xv0.x, xv0.y, xv0.z, xv0.w};
                const float xe1[4] = {xv1.x, xv1.y, xv1.z, xv1.w};
                const float we0[4] = {wv0.x, wv0.y, wv0.z, wv0.w};
                const float we1[4] = {wv1.x, wv1.y, wv1.z, wv1.w};
                #pragma unroll
                for (int e = 0; e < 4; ++e) {
                    xp[0][e] = fmaxf(xe0[e], 0.f);  xn[0][e] = fmaxf(-xe0[e], 0.f);
                    xp[1][e] = fmaxf(xe1[e], 0.f);  xn[1][e] = fmaxf(-xe1[e], 0.f);
                    wp[0][e] = fmaxf(we0[e], 0.f);  wn[0][e] = fmaxf(-we0[e], 0.f);
                    wp[1][e] = fmaxf(we1[e], 0.f);  wn[1][e] = fmaxf(-we1[e], 0.f);
                }
            }

            // per accumulator: 4 products, reduced with v_max3-friendly chains
            #pragma unroll
            for (int ii = 0; ii < 2; ++ii) {
                #pragma unroll
                for (int jj = 0; jj < 2; ++jj) {
                    float a;
                    a = acc_pp[ii][jj];
                    a = fmaxf(fmaxf(a, xp[ii][0] * wp[jj][0]), xp[ii][1] * wp[jj][1]);
                    a = fmaxf(fmaxf(a, xp[ii][2] * wp[jj][2]), xp[ii][3] * wp[jj][3]);
                    acc_pp[ii][jj] = a;

                    a = acc_nn[ii][jj];
                    a = fmaxf(fmaxf(a, xn[ii][0] * wn[jj][0]), xn[ii][1] * wn[jj][1]);
                    a = fmaxf(fmaxf(a, xn[ii][2] * wn[jj][2]), xn[ii][3] * wn[jj][3]);
                    acc_nn[ii][jj] = a;

                    a = acc_pn[ii][jj];
                    a = fmaxf(fmaxf(a, xp[ii][0] * wn[jj][0]), xp[ii][1] * wn[jj][1]);
                    a = fmaxf(fmaxf(a, xp[ii][2] * wn[jj][2]), xp[ii][3] * wn[jj][3]);
                    acc_pn[ii][jj] = a;

                    a = acc_np[ii][jj];
                    a = fmaxf(fmaxf(a, xn[ii][0] * wp[jj][0]), xn[ii][1] * wp[jj][1]);
                    a = fmaxf(fmaxf(a, xn[ii][2] * wp[jj][2]), xn[ii][3] * wp[jj][3]);
                    acc_np[ii][jj] = a;
                }
            }
        }
        __syncthreads();   // release buffer `cur` for restaging in iteration i+1
    }

    // epilogue: combine the four tropical terms + bias
    #pragma unroll
    for (int i = 0; i < 2; ++i) {
        #pragma unroll
        for (int j = 0; j < 2; ++j) {
            const int n = rowBase + r0 + i;
            const int m = colBase + c0 + j;
            const float r = acc_pp[i][j] + acc_nn[i][j]
                          - acc_pn[i][j] - acc_np[i][j] + bias[m];
            out[(size_t)n * M_DIM + m] = r;
        }
    }
}

extern "C" void kernel_launch(void* const* d_in, const int* in_sizes, int n_in,
                              void* d_out, int out_size, void* d_ws, size_t ws_size,
                              hipStream_t stream) {
    (void)in_sizes; (void)n_in; (void)out_size; (void)d_ws; (void)ws_size;
    const float* x    = (const float*)d_in[0];  // [512, 512]
    const float* wgt  = (const float*)d_in[1];  // [512, 512]
    const float* bias = (const float*)d_in[2];  // [512]
    float* out = (float*)d_out;                 // [512, 512]

    dim3 grid(M_DIM / TILE, N_DIM / TILE);      // 16 x 16 = 256 blocks
    TropixLinear_4140348473994_kernel<<<grid, 256, 0, stream>>>(x, wgt, bias, out);
}